// Gcn_31138512896565
// MI455X (gfx1250) — compile-verified
//
#include <hip/hip_runtime.h>
#include <math.h>

#define D_FEAT 128
#define LDS_STRIDE 132   // 128 + 4 pad: bank = (4n + k) mod 64 -> conflict-free wave32 ds_load_b64

typedef __attribute__((ext_vector_type(2))) float v2f;
typedef __attribute__((ext_vector_type(8))) float v8f;

// ---------- init kernels ----------
__global__ __launch_bounds__(256) void k_zero4(float4* __restrict__ p, int n4) {
    int i = blockIdx.x * blockDim.x + threadIdx.x;
    if (i < n4) p[i] = make_float4(0.f, 0.f, 0.f, 0.f);
}

__global__ __launch_bounds__(256) void k_init_deg(float* __restrict__ deg, int n) {
    int i = blockIdx.x * blockDim.x + threadIdx.x;
    if (i < n) deg[i] = 1.0f;  // self-loop contributes 1 to every node's degree
}

__global__ __launch_bounds__(256) void k_deg(const int* __restrict__ dst,
                                             float* __restrict__ deg, int ne) {
    int e = blockIdx.x * blockDim.x + threadIdx.x;
    if (e < ne) atomicAdd(&deg[dst[e]], 1.0f);
}

__global__ __launch_bounds__(256) void k_rsqrt(float* __restrict__ deg, int n) {
    int i = blockIdx.x * blockDim.x + threadIdx.x;
    if (i < n) deg[i] = rsqrtf(deg[i]);  // deg >= 1 always (self loop)
}

// ---------- fp32 WMMA GEMM: h = x @ W^T, W staged in LDS ----------
// W (64 KB) is loaded once per workgroup into padded LDS; 8 waves share it.
// Each wave grid-strides over 16-row stripes, computing all 8 column tiles.
// A 16x4 f32 layout: lane L -> row L%16; VGPR0/1 = K {0,1} (L<16) or {2,3} (L>=16).
// B 4x16 f32 layout: lane L -> col L%16; VGPR0/1 = K {0,1} (L<16) or {2,3} (L>=16).
// D 16x16 f32: VGPR r, lane L -> row r + 8*(L/16), col L%16.
__global__ __launch_bounds__(256) void k_gemm_wmma(const float* __restrict__ x,
                                                   const float* __restrict__ w,
                                                   float* __restrict__ h,
                                                   int row_tiles) {
    __shared__ float wlds[D_FEAT * LDS_STRIDE];  // 67,584 bytes

    // Cooperative stage of W: 128 rows x 32 float4 = 4096 float4 by 256 threads.
    for (int idx = threadIdx.x; idx < D_FEAT * (D_FEAT / 4); idx += 256) {
        int n  = idx >> 5;          // W row (= output column)
        int c4 = (idx & 31) << 2;   // starting k within row
        float4 v = ((const float4*)(w + (size_t)n * D_FEAT))[idx & 31];
        *(float4*)(&wlds[n * LDS_STRIDE + c4]) = v;
    }
    __syncthreads();

    const int wave  = threadIdx.x >> 5;
    const int lane  = threadIdx.x & 31;
    const int lm    = lane & 15;        // row (A) / col (B,D) within tile
    const int khalf = (lane >> 4) * 2;  // 0 or 2: K sub-offset held by this lane

    for (int stripe = blockIdx.x * 8 + wave; stripe < row_tiles;
         stripe += gridDim.x * 8) {
        const int row0 = stripe * 16;

        v8f acc[8];
#pragma unroll
        for (int t = 0; t < 8; ++t) acc[t] = (v8f)0.0f;

        const float* xrow = x + (size_t)(row0 + lm) * D_FEAT + khalf;

        for (int k = 0; k < D_FEAT; k += 4) {
            v2f a = *(const v2f*)(xrow + k);
#pragma unroll
            for (int t = 0; t < 8; ++t) {
                // B[k..][n] = W[n][k..] from LDS (conflict-free via pad)
                v2f b = *(const v2f*)(&wlds[(t * 16 + lm) * LDS_STRIDE + k + khalf]);
                acc[t] = __builtin_amdgcn_wmma_f32_16x16x4_f32(
                    false, a, false, b, (short)0, acc[t], false, false);
            }
        }

#pragma unroll
        for (int t = 0; t < 8; ++t) {
#pragma unroll
            for (int r = 0; r < 8; ++r) {
                int row = row0 + r + 8 * (lane >> 4);
                h[(size_t)row * D_FEAT + t * 16 + lm] = acc[t][r];
            }
        }
    }
}

// ---------- edge scatter: agg[dst] += norm * h[src] ----------
// One wave per edge; 128 feats / 32 lanes = one float4 per lane.
__global__ __launch_bounds__(256) void k_scatter(const float* __restrict__ h,
                                                 const int* __restrict__ src,
                                                 const int* __restrict__ dst,
                                                 const float* __restrict__ dinv,
                                                 float* __restrict__ agg, int ne) {
    int e = blockIdx.x * 8 + (threadIdx.x >> 5);
    if (e >= ne) return;
    int lane = threadIdx.x & 31;
    int s = src[e];
    int d = dst[e];
    float norm = dinv[s] * dinv[d];
    float4 v = ((const float4*)(h + (size_t)s * D_FEAT))[lane];
    float* a = agg + (size_t)d * D_FEAT + lane * 4;
    atomicAdd(a + 0, v.x * norm);
    atomicAdd(a + 1, v.y * norm);
    atomicAdd(a + 2, v.z * norm);
    atomicAdd(a + 3, v.w * norm);
}

// ---------- epilogue: out = x + relu(agg + dinv^2 * h + bias) ----------
// agg currently lives in `out`; self-loop message folded in here (no atomics).
__global__ __launch_bounds__(256) void k_final(const float* __restrict__ x,
                                               const float* __restrict__ h,
                                               const float* __restrict__ dinv,
                                               const float* __restrict__ bias,
                                               float* __restrict__ out, int n) {
    int i = blockIdx.x * blockDim.x + threadIdx.x;  // over n * 32 float4 groups
    int total4 = n * (D_FEAT / 4);
    if (i >= total4) return;
    int node = i >> 5;          // D_FEAT/4 == 32
    int f4 = (i & 31) * 4;
    float di = dinv[node];
    float self = di * di;
    float4 xa = ((const float4*)x)[i];
    float4 ha = ((const float4*)h)[i];
    float4 aa = ((const float4*)out)[i];
    float4 ba = *(const float4*)(bias + f4);
    float4 r;
    r.x = xa.x + fmaxf(fmaf(ha.x, self, aa.x) + ba.x, 0.f);
    r.y = xa.y + fmaxf(fmaf(ha.y, self, aa.y) + ba.y, 0.f);
    r.z = xa.z + fmaxf(fmaf(ha.z, self, aa.z) + ba.z, 0.f);
    r.w = xa.w + fmaxf(fmaf(ha.w, self, aa.w) + ba.w, 0.f);
    ((float4*)out)[i] = r;
}

extern "C" void kernel_launch(void* const* d_in, const int* in_sizes, int n_in,
                              void* d_out, int out_size, void* d_ws, size_t ws_size,
                              hipStream_t stream) {
    const float* x    = (const float*)d_in[0];
    const int*   edge = (const int*)d_in[1];
    const float* w    = (const float*)d_in[2];
    const float* bias = (const float*)d_in[3];
    float* out = (float*)d_out;

    const int n  = in_sizes[0] / D_FEAT;   // 100000
    const int ne = in_sizes[1] / 2;        // 600000
    const int* src = edge;                 // edge_index[0] = message sources
    const int* dst = edge + ne;            // edge_index[1] = aggregation targets

    // workspace: deg/dinv [n] floats, then h [n*128] floats (16B aligned)
    float* deg = (float*)d_ws;
    size_t hoff = ((size_t)n + 3) & ~(size_t)3;
    float* h = deg + hoff;

    const int n4 = n * (D_FEAT / 4);

    // 1) zero agg accumulator (d_out) and init degrees (self-loop = 1)
    k_zero4<<<(n4 + 255) / 256, 256, 0, stream>>>((float4*)out, n4);
    k_init_deg<<<(n + 255) / 256, 256, 0, stream>>>(deg, n);

    // 2) degree accumulation over edges
    k_deg<<<(ne + 255) / 256, 256, 0, stream>>>(dst, deg, ne);

    // 3) dinv = rsqrt(deg), in place
    k_rsqrt<<<(n + 255) / 256, 256, 0, stream>>>(deg, n);

    // 4) h = x @ W^T via fp32 WMMA; W staged once per block in LDS,
    //    each wave grid-strides over ~4 row stripes to amortize the fill.
    int row_tiles = n / 16;                       // 6250
    int gemm_blocks = (row_tiles + 8 * 4 - 1) / (8 * 4);  // ~196 blocks
    k_gemm_wmma<<<gemm_blocks, 256, 0, stream>>>(x, w, h, row_tiles);

    // 5) edge scatter with f32 atomics (one wave per edge)
    k_scatter<<<(ne + 7) / 8, 256, 0, stream>>>(h, src, dst, deg, out, ne);

    // 6) epilogue: residual + relu + bias + self-loop term
    k_final<<<(n4 + 255) / 256, 256, 0, stream>>>(x, h, deg, bias, out, n);
}